// ProductVectorQuantizer_38122129719815
// MI455X (gfx1250) — compile-verified
//
#include <hip/hip_runtime.h>
#include <hip/hip_bf16.h>
#include <math.h>

// Problem constants (from reference)
#define B_DIM   8192
#define L_DIM   4
#define K_DIM   2048
#define SUB_DIM 256
#define BETA    0.25f
#define SK_ITERS 5
#define INV_EPS (1.0f / 0.003f)
#define LOGK 7.6246189861593985f   // log(2048)
#define LOGB 9.0109133472596f      // log(8192)

typedef float v2f __attribute__((ext_vector_type(2)));
typedef float v8f __attribute__((ext_vector_type(8)));
typedef unsigned int v4u __attribute__((ext_vector_type(4)));
typedef int v4i __attribute__((ext_vector_type(4)));
typedef int v8i __attribute__((ext_vector_type(8)));

#if defined(__has_builtin)
#if __has_builtin(__builtin_amdgcn_tensor_load_to_lds) && \
    __has_builtin(__builtin_amdgcn_s_wait_tensorcnt)
#define HAVE_TDM 1
#endif
#endif
#ifndef HAVE_TDM
#define HAVE_TDM 0
#endif

// d_out layout (floats): x_q, loss, indices(float), d
#define OFF_XQ   0
#define OFF_LOSS (B_DIM * L_DIM * SUB_DIM)                 // 8388608
#define OFF_IDX  (OFF_LOSS + 1)
#define OFF_D    (OFF_IDX + B_DIM * L_DIM)                 // 8421377

// workspace layout (floats)
#define NCHUNK 16                                          // B row chunks for column LSE
#define WS_X2 0
#define WS_C2 (WS_X2 + B_DIM * L_DIM)
#define WS_R  (WS_C2 + L_DIM * K_DIM)
#define WS_C  (WS_R  + L_DIM * B_DIM)
#define WS_PM (WS_C  + L_DIM * K_DIM)
#define WS_PS (WS_PM + L_DIM * NCHUNK * K_DIM)
#define WS_LP (WS_PS + L_DIM * NCHUNK * K_DIM)

// ---------------------------------------------------------------------------
// Row-wise sum of squares: out[row] = sum(in[row][0..255]^2). One wave per row.
// ---------------------------------------------------------------------------
__global__ __launch_bounds__(256) void row_sumsq(const float* __restrict__ in,
                                                 float* __restrict__ out, int rows) {
  int wave = (int)((blockIdx.x * blockDim.x + threadIdx.x) >> 5);
  int lane = threadIdx.x & 31;
  if (wave >= rows) return;
  const float* rp = in + (size_t)wave * SUB_DIM;
  float s = 0.f;
#pragma unroll
  for (int i = 0; i < SUB_DIM / 32; ++i) {
    float v = rp[lane + i * 32];
    s += v * v;
  }
#pragma unroll
  for (int off = 16; off; off >>= 1) s += __shfl_xor(s, off, 32);
  if (lane == 0) out[wave] = s;
}

#if HAVE_TDM
// ---------------------------------------------------------------------------
// Issue one TDM 2D tile load: tile_w x tile_h elements (4B each) from global
// (row stride row_stride elements) into LDS at lds_off, with LDS padding of
// 4 dwords after every 16 dwords (matches the [128][20] padded LDS layout).
// D# bit packing per CDNA5 ISA ch. 8 (groups 0/1; 2D tensor -> groups 2/3 zero).
// This toolchain exposes the 6-arg builtin:
//   (uint32x4 g0, int32x8 g1, int32x4, int32x4, int32x8, i32 cpol)
// ---------------------------------------------------------------------------
__device__ __forceinline__ void tdm_load_tile(unsigned lds_off, const float* gptr,
                                              unsigned tile_w, unsigned tile_h,
                                              unsigned row_stride) {
  unsigned long long ga = (unsigned long long)(uintptr_t)gptr;
  v4u g0;
  g0.x = 1u;                                       // count=1, user D#, no gather
  g0.y = lds_off;                                  // lds_addr (bytes)
  g0.z = (unsigned)(ga & 0xffffffffu);             // global_addr[31:0]
  g0.w = (unsigned)((ga >> 32) & 0x01ffffffu)      // global_addr[56:32]
         | (2u << 30);                             // type = 2 ("image")
  v8i g1;
  g1[0] = (int)((2u << 16)                         // data_size = 4 bytes
              | (1u << 20)                         // pad_enable
              | (3u << 22)                         // pad_interval: 16 dwords
              | (3u << 25));                       // pad_amount: 4 dwords
  g1[1] = (int)((tile_w & 0xffffu) << 16);         // tensor_dim0[15:0] (abar=0)
  g1[2] = (int)(((tile_w >> 16) & 0xffffu) |       // tensor_dim0[31:16]
                ((tile_h & 0xffffu) << 16));       // tensor_dim1[15:0]
  g1[3] = (int)(((tile_h >> 16) & 0xffffu) |       // tensor_dim1[31:16]
                ((tile_w & 0xffffu) << 16));       // tile_dim0
  g1[4] = (int)(tile_h & 0xffffu);                 // tile_dim1 (tile_dim2 = 0)
  g1[5] = (int)row_stride;                         // tensor_dim0_stride[31:0]
  g1[6] = 0;                                       // stride0[47:32], stride1[15:0]
  g1[7] = 0;                                       // tensor_dim1_stride (unused)
  v4i gz4 = {0, 0, 0, 0};
  v8i gz8 = {0, 0, 0, 0, 0, 0, 0, 0};
  __builtin_amdgcn_tensor_load_to_lds(g0, g1, gz4, gz4, gz8, 0);
}
#endif

// ---------------------------------------------------------------------------
// Distance GEMM with fp32 WMMA:
//   d[b, l, k] = ||xs[b,l]||^2 + ||cb[l,k]||^2 - 2 * dot(xs[b,l], cb[l,k])
// Block = 256 threads (8 waves), 128(M=b) x 128(N=k) tile.
// Double-buffered LDS fed by the Tensor Data Mover (wave 0 issues the DMA,
// completion tracked with TENSORcnt); each wave owns a 32x64 region =
// 2x4 tiles of v_wmma_f32_16x16x4_f32 (32 WMMAs per 16-wide K chunk).
// ---------------------------------------------------------------------------
__global__ __launch_bounds__(256) void dist_gemm(const float* __restrict__ x,
                                                 const float* __restrict__ cb,
                                                 const float* __restrict__ x2,
                                                 const float* __restrict__ c2,
                                                 float* __restrict__ dmat) {
  __shared__ float As[2][128][20];  // 20-float row pitch: conflict-free ds_load_b64
  __shared__ float Bs[2][128][20];

  const int l  = blockIdx.z;
  const int n0 = blockIdx.x * 128;
  const int m0 = blockIdx.y * 128;
  const int tid   = threadIdx.x;
  const int lane  = tid & 31;
  const int wave  = tid >> 5;
  const int wm    = (wave >> 1) * 32;   // wave M origin in tile
  const int wn    = (wave & 1) * 64;    // wave N origin in tile
  const int half  = lane >> 4;          // 0: lanes 0-15, 1: lanes 16-31
  const int idx16 = lane & 15;

  const float* xA = x  + (size_t)m0 * (L_DIM * SUB_DIM) + l * SUB_DIM;  // row stride 1024
  const float* cB = cb + ((size_t)l * K_DIM + n0) * SUB_DIM;            // row stride 256

  v8f acc[2][4];
#pragma unroll
  for (int tm = 0; tm < 2; ++tm)
#pragma unroll
    for (int tn = 0; tn < 4; ++tn)
#pragma unroll
      for (int i = 0; i < 8; ++i) acc[tm][tn][i] = 0.f;

#if HAVE_TDM
  const unsigned ldsA0 = (unsigned)(uintptr_t)&As[0][0][0];
  const unsigned ldsA1 = (unsigned)(uintptr_t)&As[1][0][0];
  const unsigned ldsB0 = (unsigned)(uintptr_t)&Bs[0][0][0];
  const unsigned ldsB1 = (unsigned)(uintptr_t)&Bs[1][0][0];
  if (wave == 0) {  // prologue: fill buffer 0 with k-chunk 0
    tdm_load_tile(ldsA0, xA, 16, 128, L_DIM * SUB_DIM);
    tdm_load_tile(ldsB0, cB, 16, 128, SUB_DIM);
  }
#else
  const int lr = tid >> 1;       // 0..127: row this thread stages
  const int lq = (tid & 1) * 8;  // col base within 16-wide K chunk
  float4 ra0 = *(const float4*)(xA + (size_t)lr * (L_DIM * SUB_DIM) + lq);
  float4 ra1 = *(const float4*)(xA + (size_t)lr * (L_DIM * SUB_DIM) + lq + 4);
  float4 rb0 = *(const float4*)(cB + (size_t)lr * SUB_DIM + lq);
  float4 rb1 = *(const float4*)(cB + (size_t)lr * SUB_DIM + lq + 4);
#endif

  for (int step = 0; step < SUB_DIM / 16; ++step) {
    const int buf = step & 1;

#if HAVE_TDM
    if (wave == 0) __builtin_amdgcn_s_wait_tensorcnt((short)0);
    __syncthreads();  // buffer `buf` filled; prior reads of buf^1 retired
    if (step + 1 < SUB_DIM / 16 && wave == 0) {
      const int kn = (step + 1) * 16;
      tdm_load_tile(buf ? ldsA0 : ldsA1, xA + kn, 16, 128, L_DIM * SUB_DIM);
      tdm_load_tile(buf ? ldsB0 : ldsB1, cB + kn, 16, 128, SUB_DIM);
    }
#else
    __syncthreads();
    *(float4*)&As[buf][lr][lq]     = ra0;
    *(float4*)&As[buf][lr][lq + 4] = ra1;
    *(float4*)&Bs[buf][lr][lq]     = rb0;
    *(float4*)&Bs[buf][lr][lq + 4] = rb1;
    __syncthreads();
    if (step + 1 < SUB_DIM / 16) {  // prefetch next chunk into registers
      const int kn = (step + 1) * 16;
      ra0 = *(const float4*)(xA + (size_t)lr * (L_DIM * SUB_DIM) + kn + lq);
      ra1 = *(const float4*)(xA + (size_t)lr * (L_DIM * SUB_DIM) + kn + lq + 4);
      rb0 = *(const float4*)(cB + (size_t)lr * SUB_DIM + kn + lq);
      rb1 = *(const float4*)(cB + (size_t)lr * SUB_DIM + kn + lq + 4);
    }
#endif

#pragma unroll
    for (int kk = 0; kk < 4; ++kk) {
      const int c = kk * 4 + half * 2;  // A/B fragment: VGPR j <-> K = 2*half + j
      v2f afr[2], bfr[4];
#pragma unroll
      for (int tm = 0; tm < 2; ++tm)
        afr[tm] = *(const v2f*)&As[buf][wm + tm * 16 + idx16][c];
#pragma unroll
      for (int tn = 0; tn < 4; ++tn)
        bfr[tn] = *(const v2f*)&Bs[buf][wn + tn * 16 + idx16][c];
#pragma unroll
      for (int tm = 0; tm < 2; ++tm)
#pragma unroll
        for (int tn = 0; tn < 4; ++tn)
          acc[tm][tn] = __builtin_amdgcn_wmma_f32_16x16x4_f32(
              false, afr[tm], false, bfr[tn], (short)0, acc[tm][tn], false, false);
    }
  }

  // Epilogue: d = |x|^2 + |c|^2 - 2*acc.  C/D layout: VGPR i <-> M = i + 8*half.
#pragma unroll
  for (int tm = 0; tm < 2; ++tm)
#pragma unroll
    for (int tn = 0; tn < 4; ++tn)
#pragma unroll
      for (int i = 0; i < 8; ++i) {
        const int gm = m0 + wm + tm * 16 + i + half * 8;
        const int gn = n0 + wn + tn * 16 + idx16;
        const float v = x2[gm * L_DIM + l] + c2[l * K_DIM + gn] - 2.0f * acc[tm][tn][i];
        dmat[(size_t)gm * (L_DIM * K_DIM) + l * K_DIM + gn] = v;
      }
}

// ---------------------------------------------------------------------------
// Sinkhorn column pass: partial online LSE over a 512-row chunk per column k.
//   v = -d/eps + r[b]   (r skipped on first iteration)
// ---------------------------------------------------------------------------
__global__ __launch_bounds__(256) void sk_colpass(const float* __restrict__ dmat,
                                                  const float* __restrict__ r,
                                                  float* __restrict__ pm,
                                                  float* __restrict__ ps, int useR) {
  const int l = blockIdx.z;
  const int k = blockIdx.x * 256 + threadIdx.x;
  const int chunk = blockIdx.y;
  const int rows = B_DIM / NCHUNK;  // 512
  const float* dl = dmat + (size_t)chunk * rows * (L_DIM * K_DIM) + l * K_DIM + k;
  const float* rl = r + l * B_DIM + chunk * rows;
  float m = -INFINITY, s = 0.f;
  for (int b = 0; b < rows; ++b) {
    float v = dl[(size_t)b * (L_DIM * K_DIM)] * (-INV_EPS);
    if (useR) v += rl[b];
    const float nm = fmaxf(m, v);
    s = s * __expf(m - nm) + __expf(v - nm);
    m = nm;
  }
  pm[(l * NCHUNK + chunk) * K_DIM + k] = m;
  ps[(l * NCHUNK + chunk) * K_DIM + k] = s;
}

// Merge column partials -> c[k] = -lse_b(-d/eps + r) - log K
__global__ __launch_bounds__(256) void sk_colmerge(const float* __restrict__ pm,
                                                   const float* __restrict__ ps,
                                                   float* __restrict__ c) {
  const int t = blockIdx.x * 256 + threadIdx.x;  // 0 .. L*K-1
  const int l = t / K_DIM, k = t % K_DIM;
  float m = -INFINITY;
#pragma unroll
  for (int j = 0; j < NCHUNK; ++j) m = fmaxf(m, pm[(l * NCHUNK + j) * K_DIM + k]);
  float s = 0.f;
#pragma unroll
  for (int j = 0; j < NCHUNK; ++j)
    s += ps[(l * NCHUNK + j) * K_DIM + k] * __expf(pm[(l * NCHUNK + j) * K_DIM + k] - m);
  c[l * K_DIM + k] = -(m + __logf(s)) - LOGK;
}

// ---------------------------------------------------------------------------
// Sinkhorn row pass: r[b] = -lse_k(-d/eps + c[k]) - log B.  One wave per row.
// ---------------------------------------------------------------------------
__global__ __launch_bounds__(256) void sk_rowpass(const float* __restrict__ dmat,
                                                  const float* __restrict__ c,
                                                  float* __restrict__ r) {
  const int l = blockIdx.y;
  const int wave = threadIdx.x >> 5, lane = threadIdx.x & 31;
  const int b = blockIdx.x * 8 + wave;
  const float* dl = dmat + (size_t)b * (L_DIM * K_DIM) + l * K_DIM;
  const float* cl = c + l * K_DIM;
  float m = -INFINITY, s = 0.f;
  for (int k = lane; k < K_DIM; k += 32) {
    const float v = dl[k] * (-INV_EPS) + cl[k];
    const float nm = fmaxf(m, v);
    s = s * __expf(m - nm) + __expf(v - nm);
    m = nm;
  }
#pragma unroll
  for (int off = 16; off; off >>= 1) {
    const float om = __shfl_xor(m, off, 32);
    const float os = __shfl_xor(s, off, 32);
    const float nm = fmaxf(m, om);
    s = s * __expf(m - nm) + os * __expf(om - nm);
    m = nm;
  }
  if (lane == 0) r[l * B_DIM + b] = -(m + __logf(s)) - LOGB;
}

// ---------------------------------------------------------------------------
// Finalize: argmax_k(c[k] - d/eps) per row, gather codebook row -> x_q,
// write index, per-block partial of sum((xq - xs)^2).
// ---------------------------------------------------------------------------
__global__ __launch_bounds__(256) void finalize(const float* __restrict__ dmat,
                                                const float* __restrict__ c,
                                                const float* __restrict__ x,
                                                const float* __restrict__ cb,
                                                float* __restrict__ out,
                                                float* __restrict__ lossp) {
  __shared__ float red[8];
  const int l = blockIdx.y;
  const int wave = threadIdx.x >> 5, lane = threadIdx.x & 31;
  const int b = blockIdx.x * 8 + wave;
  const float* dl = dmat + (size_t)b * (L_DIM * K_DIM) + l * K_DIM;
  const float* cl = c + l * K_DIM;

  float bv = -INFINITY;
  int bi = 0;
  for (int k = lane; k < K_DIM; k += 32) {
    const float v = cl[k] - dl[k] * INV_EPS;
    if (v > bv) { bv = v; bi = k; }
  }
#pragma unroll
  for (int off = 16; off; off >>= 1) {
    const float ov = __shfl_xor(bv, off, 32);
    const int   oi = __shfl_xor(bi, off, 32);
    if (ov > bv || (ov == bv && oi < bi)) { bv = ov; bi = oi; }
  }
  if (lane == 0) out[OFF_IDX + b * L_DIM + l] = (float)bi;

  const float* crow = cb + ((size_t)l * K_DIM + bi) * SUB_DIM;
  const float* xrow = x + (size_t)b * (L_DIM * SUB_DIM) + l * SUB_DIM;
  float* q = out + OFF_XQ + (size_t)b * (L_DIM * SUB_DIM) + l * SUB_DIM;
  float se = 0.f;
#pragma unroll
  for (int i = 0; i < SUB_DIM / 32; ++i) {
    const float cv = crow[lane + i * 32];
    const float xv = xrow[lane + i * 32];
    q[lane + i * 32] = cv;  // straight-through forward value == codeword
    const float dlt = cv - xv;
    se += dlt * dlt;
  }
#pragma unroll
  for (int off = 16; off; off >>= 1) se += __shfl_xor(se, off, 32);
  if (lane == 0) red[wave] = se;
  __syncthreads();
  if (threadIdx.x == 0) {
    float t = 0.f;
    for (int w = 0; w < 8; ++w) t += red[w];
    lossp[blockIdx.y * gridDim.x + blockIdx.x] = t;
  }
}

// Deterministic final loss reduction (no float atomics)
__global__ __launch_bounds__(256) void loss_reduce(const float* __restrict__ lossp,
                                                   float* __restrict__ out, int n) {
  __shared__ float red[256];
  float s = 0.f;
  for (int i = threadIdx.x; i < n; i += 256) s += lossp[i];
  red[threadIdx.x] = s;
  __syncthreads();
  for (int off = 128; off; off >>= 1) {
    if ((int)threadIdx.x < off) red[threadIdx.x] += red[threadIdx.x + off];
    __syncthreads();
  }
  if (threadIdx.x == 0)
    out[OFF_LOSS] = red[0] * ((1.0f + BETA) / (float)((size_t)B_DIM * L_DIM * SUB_DIM));
}

// ---------------------------------------------------------------------------
extern "C" void kernel_launch(void* const* d_in, const int* in_sizes, int n_in,
                              void* d_out, int out_size, void* d_ws, size_t ws_size,
                              hipStream_t stream) {
  (void)in_sizes; (void)n_in; (void)out_size; (void)ws_size;
  const float* x  = (const float*)d_in[0];
  const float* cb = (const float*)d_in[1];
  float* out = (float*)d_out;
  float* ws  = (float*)d_ws;

  float* x2 = ws + WS_X2;
  float* c2 = ws + WS_C2;
  float* r  = ws + WS_R;
  float* c  = ws + WS_C;
  float* pm = ws + WS_PM;
  float* ps = ws + WS_PS;
  float* lp = ws + WS_LP;
  float* dmat = out + OFF_D;

  row_sumsq<<<dim3(B_DIM * L_DIM / 8), 256, 0, stream>>>(x, x2, B_DIM * L_DIM);
  row_sumsq<<<dim3(L_DIM * K_DIM / 8), 256, 0, stream>>>(cb, c2, L_DIM * K_DIM);

  dist_gemm<<<dim3(K_DIM / 128, B_DIM / 128, L_DIM), 256, 0, stream>>>(x, cb, x2, c2, dmat);

  for (int it = 0; it < SK_ITERS; ++it) {
    sk_colpass<<<dim3(K_DIM / 256, NCHUNK, L_DIM), 256, 0, stream>>>(dmat, r, pm, ps, it > 0);
    sk_colmerge<<<dim3(L_DIM * K_DIM / 256), 256, 0, stream>>>(pm, ps, c);
    if (it < SK_ITERS - 1)
      sk_rowpass<<<dim3(B_DIM / 8, L_DIM), 256, 0, stream>>>(dmat, c, r);
  }

  finalize<<<dim3(B_DIM / 8, L_DIM), 256, 0, stream>>>(dmat, c, x, cb, out, lp);
  loss_reduce<<<1, 256, 0, stream>>>(lp, out, L_DIM * (B_DIM / 8));
}